// UMPS_41695542509690
// MI455X (gfx1250) — compile-verified
//
#include <hip/hip_runtime.h>
#include <hip/hip_bf16.h>

// UMPS chain contraction for MI455X (gfx1250, wave32, WMMA).
// B=512, L=128, FEAT=33, D=256, O=16.
//
// Per step: v_new[b,e] = sum_{d,f} v[b,d]*x[b,f]*core[d,f,e]
//   == GEMM  Z(16b x 8448k) @ W(8448k x 256n), Z built on the fly in regs.
// 64 WGs = 32 batch tiles x {left,right half}; v lives in LDS (double
// buffered, row pad 260 => conflict-free reads). Each of 8 waves owns 32
// output columns. W is repacked once into a WMMA-native layout so each
// WMMA pair needs exactly one global_load_b128 for its B operands, with
// 4KB-strided blocks (immediate-offset addressing in the unrolled loop).
// Matrix op: V_WMMA_F32_16X16X4_F32 (full f32) for reference-grade
// precision over the 128-step chain.

#define BB   512
#define LL   128
#define FEAT 33
#define DD   256
#define OO   16
#define VPAD 260   // 256+4: lane banks (4m+k) mod 64 -> all 64 banks once
#define XPAD 34
#define WSIDE (FEAT * DD * DD)   // 2162688 floats per packed side

typedef __attribute__((ext_vector_type(2))) float v2f;
typedef __attribute__((ext_vector_type(4))) float v4f;
typedef __attribute__((ext_vector_type(8))) float v8f;

// ---------------------------------------------------------------------------
// Kernel 1: pack core into WMMA-friendly B layout, both orientations.
//   Wp[side][f][k4][h][w][ln][c], c=0..3:
//     k = k4*4 + 2h + (c&1),  n = w*32 + (c>>1)*16 + ln
//   side 0 (left scan):  element = core[d=k][f][e=n]
//   side 1 (right scan): element = core[d=n][f][e=k]
// A lane (h, ln) of wave w then loads its 2 B fragments (n and n+16 tiles,
// K pair) as ONE b128 at offset h*512 + w*64 + ln*4 within a 1024-float
// (f,k4) block; consecutive k4 blocks are 4KB apart.
// ---------------------------------------------------------------------------
__global__ void umps_pack(const float* __restrict__ core,
                          float* __restrict__ Wp) {
  const int side = blockIdx.y;
  const int idx  = blockIdx.x * 256 + threadIdx.x;   // < FEAT*64*1024
  const int c  = idx & 3;
  const int ln = (idx >> 2) & 15;
  const int w  = (idx >> 6) & 7;
  const int h  = (idx >> 9) & 1;
  const int k4 = (idx >> 10) & 63;
  const int f  = idx >> 16;
  const int k = k4 * 4 + 2 * h + (c & 1);
  const int n = w * 32 + (c >> 1) * 16 + ln;
  const int d = side ? n : k;
  const int e = side ? k : n;
  Wp[side * WSIDE + idx] = core[(d * FEAT + f) * DD + e];
}

// ---------------------------------------------------------------------------
// Kernel 2: the chain. grid = 64 (btile*2 + side), block = 256 (8 waves).
// ---------------------------------------------------------------------------
__global__ __launch_bounds__(256, 1) void umps_chain(
    const float* __restrict__ inputs,  // (B,L,FEAT)
    const float* __restrict__ Wp,      // packed, 2 sides
    const float* __restrict__ alpha,
    const float* __restrict__ omega,
    float* __restrict__ leftOut,       // (B,D)
    float* __restrict__ rightOut)      // (B,D)
{
  __shared__ float vbuf[2][16 * VPAD];
  __shared__ float xs[16 * XPAD];

  const int tid  = threadIdx.x;
  const int lane = tid & 31;
  const int wave = tid >> 5;
  const int h    = lane >> 4;   // lane half: K sub-pair in A/B frags
  const int ln   = lane & 15;   // A row (batch) / B,C column
  const int side  = blockIdx.x & 1;
  const int btile = blockIdx.x >> 1;
  const int b0    = btile * 16;

  const float* W   = Wp + side * WSIDE;
  const float* bnd = side ? omega : alpha;
  float* vout      = side ? rightOut : leftOut;

  // init boundary vector, replicated across the 16 batch rows
  for (int i = tid; i < 16 * DD; i += 256) {
    int r = i >> 8, d = i & 255;
    vbuf[0][r * VPAD + d] = bnd[d];
  }
  int cur = 0;
  const int n0   = wave * 32;            // this wave's output-column base
  const int m    = ln;                   // this lane's A-matrix row
  const int lofs = h * 512 + wave * 64 + ln * 4;  // lane offset in W block

  for (int s = 0; s < 64; ++s) {
    __syncthreads();
    const int pos = side ? (LL - 1 - s) : s;
    for (int i = tid; i < 16 * FEAT; i += 256) {
      int r = i / FEAT, f = i - r * FEAT;
      xs[r * XPAD + f] = inputs[((b0 + r) * LL + pos) * FEAT + f];
    }
    __syncthreads();

    const float* vc = vbuf[cur];
    v8f acc0a = {}, acc0b = {}, acc1a = {}, acc1b = {};

    for (int kc = 0; kc < 8; ++kc) {            // K chunks of 32
      const int kb = kc * 32;
      v2f vr[8];                                // v[m][kb..kb+31], this half
      #pragma unroll
      for (int j = 0; j < 8; ++j)
        vr[j] = *(const v2f*)(vc + m * VPAD + kb + j * 4 + 2 * h);

      for (int f = 0; f < FEAT; ++f) {
        const float xf = xs[m * XPAD + f];
        const float* Wb = W + f * (64 * 1024) + kc * (8 * 1024) + lofs;
        #pragma unroll
        for (int j = 0; j < 8; ++j) {
          v2f a;                                 // z = x[b,f]*v[b,k]
          a[0] = xf * vr[j][0];
          a[1] = xf * vr[j][1];
          const v4f bq = *(const v4f*)(Wb + j * 1024);   // both B frags
          v2f bA; bA[0] = bq[0]; bA[1] = bq[1];
          v2f bC; bC[0] = bq[2]; bC[1] = bq[3];
          if (j & 1) {
            acc0b = __builtin_amdgcn_wmma_f32_16x16x4_f32(
                false, a, false, bA, (short)0, acc0b, false, false);
            acc1b = __builtin_amdgcn_wmma_f32_16x16x4_f32(
                false, a, false, bC, (short)0, acc1b, false, false);
          } else {
            acc0a = __builtin_amdgcn_wmma_f32_16x16x4_f32(
                false, a, false, bA, (short)0, acc0a, false, false);
            acc1a = __builtin_amdgcn_wmma_f32_16x16x4_f32(
                false, a, false, bC, (short)0, acc1a, false, false);
          }
        }
      }
    }
    const v8f acc0 = acc0a + acc0b;
    const v8f acc1 = acc1a + acc1b;

    // scatter D tile into v_next (C/D layout: reg i -> row i+8h, col ln)
    const int nxt = cur ^ 1;
    float* vn = vbuf[nxt];
    #pragma unroll
    for (int i = 0; i < 8; ++i) {
      const int r = i + 8 * h;
      vn[r * VPAD + n0 + ln]      = acc0[i];
      vn[r * VPAD + n0 + 16 + ln] = acc1[i];
      if (s == 63) {
        vout[(b0 + r) * DD + n0 + ln]      = acc0[i];
        vout[(b0 + r) * DD + n0 + 16 + ln] = acc1[i];
      }
    }
    cur = nxt;
  }
}

// ---------------------------------------------------------------------------
// Kernel 3: out[b,o] = sum_{d,e} left[b,d] * ocore[d,o,e] * right[b,e]
// as GEMM (16b x 256d) @ (256d x 4096(o,e)) with the e-fold via ds_add_f32.
// grid = 32 (16 batches each), block = 256.
// ---------------------------------------------------------------------------
__global__ __launch_bounds__(256, 1) void umps_combine(
    const float* __restrict__ leftB,
    const float* __restrict__ rightB,
    const float* __restrict__ ocore,   // (D,O,D): addr d*4096 + o*256 + e
    float* __restrict__ out)           // (B,O)
{
  __shared__ float Ls[16 * VPAD];
  __shared__ float Rs[16 * VPAD];
  __shared__ float red[16 * OO];

  const int tid  = threadIdx.x;
  const int lane = tid & 31;
  const int wave = tid >> 5;
  const int h    = lane >> 4;
  const int ln   = lane & 15;
  const int b0   = blockIdx.x * 16;

  for (int i = tid; i < 16 * DD; i += 256) {
    int r = i >> 8, d = i & 255;
    Ls[r * VPAD + d] = leftB[(b0 + r) * DD + d];
    Rs[r * VPAD + d] = rightB[(b0 + r) * DD + d];
  }
  if (tid < 16 * OO) red[tid] = 0.0f;
  __syncthreads();

  const int m = ln;
  for (int t = 0; t < 32; ++t) {       // wave's 512 of 4096 (o,e) columns
    const int nbase = wave * 512 + t * 16;      // single o per tile
    const int o     = nbase >> 8;
    const int ebase = nbase & 255;
    v8f acc = {};
    const float* lrow = Ls + m * VPAD + 2 * h;
    const float* Wn   = ocore + nbase + ln;
    #pragma unroll 4
    for (int kd = 0; kd < DD; kd += 4) {
      v2f a = *(const v2f*)(lrow + kd);
      const int ki = (kd + 2 * h) * (OO * DD);
      v2f b;
      b[0] = Wn[ki];
      b[1] = Wn[ki + OO * DD];
      acc = __builtin_amdgcn_wmma_f32_16x16x4_f32(
          false, a, false, b, (short)0, acc, false, false);
    }
    #pragma unroll
    for (int i = 0; i < 8; ++i) {
      const int r = i + 8 * h;
      const float val = acc[i] * Rs[r * VPAD + ebase + ln];
      atomicAdd(&red[r * OO + o], val);         // ds_add_f32
    }
  }
  __syncthreads();
  if (tid < 16 * OO) {
    const int bb = tid >> 4, o = tid & 15;
    out[(b0 + bb) * OO + o] = red[bb * OO + o];
  }
}

// ---------------------------------------------------------------------------
extern "C" void kernel_launch(void* const* d_in, const int* in_sizes, int n_in,
                              void* d_out, int out_size, void* d_ws, size_t ws_size,
                              hipStream_t stream) {
  const float* inputs = (const float*)d_in[0];   // B*L*FEAT
  const float* core   = (const float*)d_in[1];   // D*FEAT*D
  const float* alpha  = (const float*)d_in[2];   // D
  const float* omega  = (const float*)d_in[3];   // D
  const float* ocore  = (const float*)d_in[4];   // D*O*D
  float* out = (float*)d_out;                    // B*O

  // workspace layout (floats): Wp[2 sides] | left | right  (~18.4 MB)
  float* Wp     = (float*)d_ws;
  float* leftB  = Wp + 2 * WSIDE;
  float* rightB = leftB + BB * DD;

  dim3 pgrid(WSIDE / 256, 2);
  umps_pack<<<pgrid, 256, 0, stream>>>(core, Wp);
  umps_chain<<<(BB / 16) * 2, 256, 0, stream>>>(inputs, Wp, alpha, omega,
                                                leftB, rightB);
  umps_combine<<<BB / 16, 256, 0, stream>>>(leftB, rightB, ocore, out);
}